// RGCNHetero_3908420239951
// MI455X (gfx1250) — compile-verified
//
#include <hip/hip_runtime.h>

typedef __attribute__((ext_vector_type(2))) float v2f;
typedef __attribute__((ext_vector_type(8))) float v8f;

#define NODES   100000
#define RELS    8
#define EDGES   100000
#define DFEAT   64

#define EDGES_PER_WG   128
#define THREADS        256
#define PITCH          68   // LDS row pitch in floats (64 + 4 pad -> conflict-free fragment reads)

__global__ __launch_bounds__(THREADS)
void rgcn_zero_kernel(float* __restrict__ out, int n4) {
    int i = blockIdx.x * blockDim.x + threadIdx.x;
    if (i < n4) ((float4*)out)[i] = make_float4(0.f, 0.f, 0.f, 0.f);
}

__global__ __launch_bounds__(THREADS)
void rgcn_edge_wmma_kernel(const float* __restrict__ feat,
                           const float* __restrict__ weight,
                           const int*   __restrict__ esrc,
                           const int*   __restrict__ edst,
                           float*       __restrict__ out) {
    __shared__ float sWt[DFEAT * PITCH];          // W^T for this relation: sWt[n][k]
    __shared__ float sA [EDGES_PER_WG * PITCH];   // gathered source feature rows
    __shared__ int   sSrc[EDGES_PER_WG];
    __shared__ int   sDst[EDGES_PER_WG];

    const int tid   = threadIdx.x;
    const int rel   = blockIdx.y;
    const int ebase = blockIdx.x * EDGES_PER_WG;

    // Stage edge indices (tail edges: clamp src for safe loads, dst=-1 disables scatter).
    if (tid < EDGES_PER_WG) {
        int e  = ebase + tid;
        int ec = (e < EDGES) ? e : (EDGES - 1);
        sSrc[tid] = esrc[rel * EDGES + ec];
        sDst[tid] = (e < EDGES) ? edst[rel * EDGES + e] : -1;
    }
    __syncthreads();

    // Stage W^T: sWt[n*PITCH + k] = W[rel][k][n]  (reads coalesced over n)
    for (int i = tid; i < DFEAT * DFEAT; i += THREADS) {
        int k = i >> 6, n = i & 63;
        sWt[n * PITCH + k] = weight[((size_t)rel * DFEAT + k) * DFEAT + n];
    }
    // Gather A rows: 16 lanes x float4 cover one 256B feature row (coalesced per row).
    for (int i = tid; i < EDGES_PER_WG * 16; i += THREADS) {
        int e = i >> 4;
        int g = (i & 15) << 2;
        const float4 v = *(const float4*)&feat[(size_t)sSrc[e] * DFEAT + g];
        *(float4*)&sA[e * PITCH + g] = v;
    }
    __syncthreads();

    // Each wave (wave32) computes a 16-edge x 64-out tile with fp32 WMMA.
    const int wave = tid >> 5;
    const int lane = tid & 31;
    const int ln   = lane & 15;
    const int half = lane >> 4;   // 0: K/M lower half, 1: upper half per ISA layout

    const float* aRow = &sA[(wave * 16 + ln) * PITCH];
    v8f acc0 = {}, acc1 = {}, acc2 = {}, acc3 = {};

    #pragma unroll
    for (int k0 = 0; k0 < DFEAT; k0 += 4) {
        const int ko = k0 + 2 * half;   // A VGPR0/1 hold K={k0,k0+1} (lanes 0-15) / K={k0+2,k0+3} (lanes 16-31)
        v2f a  = *(const v2f*)&aRow[ko];
        v2f b0 = *(const v2f*)&sWt[( 0 + ln) * PITCH + ko];
        v2f b1 = *(const v2f*)&sWt[(16 + ln) * PITCH + ko];
        v2f b2 = *(const v2f*)&sWt[(32 + ln) * PITCH + ko];
        v2f b3 = *(const v2f*)&sWt[(48 + ln) * PITCH + ko];
        acc0 = __builtin_amdgcn_wmma_f32_16x16x4_f32(false, a, false, b0, (short)0, acc0, false, false);
        acc1 = __builtin_amdgcn_wmma_f32_16x16x4_f32(false, a, false, b1, (short)0, acc1, false, false);
        acc2 = __builtin_amdgcn_wmma_f32_16x16x4_f32(false, a, false, b2, (short)0, acc2, false, false);
        acc3 = __builtin_amdgcn_wmma_f32_16x16x4_f32(false, a, false, b3, (short)0, acc3, false, false);
    }

    // Scatter-add messages: C/D VGPR i = row (i + 8*half), column j*16 + ln.
    #pragma unroll
    for (int i = 0; i < 8; ++i) {
        int m   = wave * 16 + i + 8 * half;
        int dst = sDst[m];
        if (dst >= 0) {
            float* o = &out[(size_t)dst * DFEAT + ln];
            atomicAdd(o +  0, acc0[i]);
            atomicAdd(o + 16, acc1[i]);
            atomicAdd(o + 32, acc2[i]);
            atomicAdd(o + 48, acc3[i]);
        }
    }
}

extern "C" void kernel_launch(void* const* d_in, const int* in_sizes, int n_in,
                              void* d_out, int out_size, void* d_ws, size_t ws_size,
                              hipStream_t stream) {
    (void)in_sizes; (void)n_in; (void)d_ws; (void)ws_size;

    const float* feat   = (const float*)d_in[0];
    const float* weight = (const float*)d_in[1];
    const int*   esrc   = (const int*)d_in[2];
    const int*   edst   = (const int*)d_in[3];
    float*       out    = (float*)d_out;

    // Zero the accumulator output every launch (atomic scatter-add follows).
    int n4 = out_size / 4;
    rgcn_zero_kernel<<<(n4 + THREADS - 1) / THREADS, THREADS, 0, stream>>>(out, n4);

    dim3 grid((EDGES + EDGES_PER_WG - 1) / EDGES_PER_WG, RELS);
    rgcn_edge_wmma_kernel<<<grid, THREADS, 0, stream>>>(feat, weight, esrc, edst, out);
}